// VectorQuantizer_59691455480151
// MI455X (gfx1250) — compile-verified
//
#include <hip/hip_runtime.h>
#include <hip/hip_bf16.h>

// ---------------------------------------------------------------------------
// VQ-VAE vector quantizer for MI455X (gfx1250, wave32)
//   - distances via chained v_wmma_f32_16x16x4_f32 (exact f32 argmin)
//   - tile staging via global_load_async_to_lds_b128 (ASYNCcnt pipe, no VGPRs)
//   inputs:   [16,2048,512] f32  -> N=32768 rows, D=512
//   codebook: [8192,512]    f32  -> K=8192 codes
//   out (concat, f32): quantized_st [N*D], commitment_loss [1], indices [N]
// ---------------------------------------------------------------------------

#define DIMS       512
#define KCODES     8192
#define NROWS      32768
#define M_TILE     64          // x rows per workgroup
#define N_TILE     32          // codes per LDS tile (double buffered)
#define LSTRIDE    516         // 512 + 4 floats pad -> conflict-free 64-bank LDS
#define BETA_F     0.01

typedef float v2f __attribute__((ext_vector_type(2)));
typedef float v8f __attribute__((ext_vector_type(8)));

// Async global->LDS copy of an nRows x 512 f32 tile into padded-row LDS.
// GVS addressing: saddr = uniform tile base, vaddr = per-lane byte offset,
// vdst = per-lane LDS byte offset (low 32 bits of the generic shared pointer).
__device__ __forceinline__
void async_copy_tile(float* ldsDst, const float* gSrc, int nRows, int tid) {
    for (int li = tid; li < nRows * (DIMS / 4); li += 256) {
        const int r  = li >> 7;                 // DIMS/4 == 128
        const int c4 = li & 127;
        const unsigned ldsOff = (unsigned)(uintptr_t)(ldsDst + r * LSTRIDE + c4 * 4);
        const unsigned gOff   = (unsigned)((r * DIMS + c4 * 4) * sizeof(float));
        asm volatile("global_load_async_to_lds_b128 %0, %1, %2"
                     :
                     : "v"(ldsOff), "v"(gOff), "s"(gSrc)
                     : "memory");
    }
}

__device__ __forceinline__ void wait_async_done() {
    asm volatile("s_wait_asynccnt 0x0" ::: "memory");
}

// -------------------------------- kernel 1: codebook squared norms ----------
__global__ __launch_bounds__(128)
void vq_norms_kernel(const float* __restrict__ cb, float* __restrict__ enorm) {
    __shared__ float red[128];
    const int k = blockIdx.x;
    const float* row = cb + (size_t)k * DIMS;
    float s = 0.f;
    for (int d = threadIdx.x; d < DIMS; d += 128) {
        float v = row[d];
        s += v * v;
    }
    red[threadIdx.x] = s;
    __syncthreads();
    for (int off = 64; off > 0; off >>= 1) {
        if (threadIdx.x < off) red[threadIdx.x] += red[threadIdx.x + off];
        __syncthreads();
    }
    if (threadIdx.x == 0) enorm[k] = red[0];
}

// -------------------------------- kernel 2: fused distance + argmin ---------
// 256 threads = 8 waves. Wave w -> (mi = w>>1) in M, (ni = w&1) in N.
// Each wave owns a 16x16 score tile, accumulated over D=512 with 128 chained
// v_wmma_f32_16x16x4_f32 ops reading A (x tile) and B (codebook tile) from LDS.
__global__ __launch_bounds__(256)
void vq_argmin_kernel(const float* __restrict__ x,
                      const float* __restrict__ cb,
                      const float* __restrict__ enorm,
                      int* __restrict__ outIdx) {
    extern __shared__ float smem[];
    float* xs   = smem;                              // M_TILE * LSTRIDE
    float* cbs  = xs  + M_TILE * LSTRIDE;            // 2 * N_TILE * LSTRIDE
    float* sMin = cbs + 2 * N_TILE * LSTRIDE;        // M_TILE * N_TILE
    int*   sIdx = (int*)(sMin + M_TILE * N_TILE);    // M_TILE * N_TILE

    const int tid  = threadIdx.x;
    const int lane = tid & 31;
    const int wave = tid >> 5;
    const int mi   = wave >> 1;      // 0..3 : which 16-row M sub-tile
    const int ni   = wave & 1;       // 0..1 : which 16-code N sub-tile
    const int r16  = lane & 15;
    const int hi   = lane >> 4;      // lane half selects K pair (ISA 7.12.2)

    const int rowBase = blockIdx.x * M_TILE;

    // Stage the 64x512 x tile and codebook tile 0 through the async pipe.
    async_copy_tile(xs, x + (size_t)rowBase * DIMS, M_TILE, tid);
    async_copy_tile(cbs, cb, N_TILE, tid);
    wait_async_done();
    __syncthreads();

    float minv[8];
    int   mini[8];
#pragma unroll
    for (int i = 0; i < 8; ++i) { minv[i] = 3.4e38f; mini[i] = 0; }

    const int NTILES = KCODES / N_TILE;   // 256
    for (int t = 0; t < NTILES; ++t) {
        // Async-fill the other buffer while this one is consumed by WMMAs.
        if (t + 1 < NTILES) {
            async_copy_tile(cbs + ((t + 1) & 1) * N_TILE * LSTRIDE,
                            cb + (size_t)(t + 1) * N_TILE * DIMS, N_TILE, tid);
        }

        const float* abuf = xs + (mi * 16 + r16) * LSTRIDE;
        const float* bbuf = cbs + (t & 1) * N_TILE * LSTRIDE
                                + (ni * 16 + r16) * LSTRIDE;
        const float  en   = enorm[t * N_TILE + ni * 16 + r16];

        // D-loop: 128 chained f32 WMMAs, K=4 each.
        // A 16x4 layout: lanes0-15 {K0,K1}, lanes16-31 {K2,K3} -> one b64 load.
        v8f c = {};
#pragma unroll 8
        for (int kk = 0; kk < DIMS / 4; ++kk) {
            const v2f a = *(const v2f*)(abuf + kk * 4 + 2 * hi);
            const v2f b = *(const v2f*)(bbuf + kk * 4 + 2 * hi);
            c = __builtin_amdgcn_wmma_f32_16x16x4_f32(
                    false, a, false, b, (short)0, c, false, false);
        }

        // score = ||e||^2 - 2*dot  (||x||^2 constant per row, dropped)
        const int code = t * N_TILE + ni * 16 + r16;
#pragma unroll
        for (int i = 0; i < 8; ++i) {
            const float s = en - 2.0f * c[i];
            if (s < minv[i]) { minv[i] = s; mini[i] = code; }
        }

        wait_async_done();   // next buffer's LDS writes complete...
        __syncthreads();     // ...and published to all waves
    }

    // Spill per-lane running minima: C layout VGPR i -> M = i + 8*hi.
#pragma unroll
    for (int i = 0; i < 8; ++i) {
        const int row = mi * 16 + hi * 8 + i;
        const int col = ni * 16 + r16;
        sMin[row * N_TILE + col] = minv[i];
        sIdx[row * N_TILE + col] = mini[i];
    }
    __syncthreads();

    // One thread per row: reduce 32 columns (ties -> lowest index, like argmin).
    if (tid < M_TILE) {
        float bv = sMin[tid * N_TILE];
        int   bi = sIdx[tid * N_TILE];
        for (int c2 = 1; c2 < N_TILE; ++c2) {
            const float v  = sMin[tid * N_TILE + c2];
            const int   ix = sIdx[tid * N_TILE + c2];
            if (v < bv || (v == bv && ix < bi)) { bv = v; bi = ix; }
        }
        outIdx[rowBase + tid] = bi;
    }
}

// -------------------------------- kernel 3: gather + loss partials ----------
__global__ __launch_bounds__(128)
void vq_gather_kernel(const float* __restrict__ x,
                      const float* __restrict__ cb,
                      const int* __restrict__ idx,
                      float* __restrict__ outQ,
                      float* __restrict__ outIdxF,
                      double* __restrict__ acc) {
    __shared__ float red[128];
    const int row = blockIdx.x;
    const int k   = idx[row];
    const float* q  = cb + (size_t)k   * DIMS;
    const float* xr = x  + (size_t)row * DIMS;
    float s = 0.f;
    for (int d = threadIdx.x; d < DIMS; d += 128) {
        const float qv = q[d];
        const float xv = xr[d];
        outQ[(size_t)row * DIMS + d] = qv;       // straight-through fwd == q
        const float df = qv - xv;
        s += df * df;
    }
    red[threadIdx.x] = s;
    __syncthreads();
    for (int off = 64; off > 0; off >>= 1) {
        if (threadIdx.x < off) red[threadIdx.x] += red[threadIdx.x + off];
        __syncthreads();
    }
    if (threadIdx.x == 0) {
        atomicAdd(acc, (double)red[0]);
        outIdxF[row] = (float)k;
    }
}

// -------------------------------- kernel 4: finalize loss -------------------
__global__ void vq_finalize_kernel(const double* __restrict__ acc,
                                   float* __restrict__ outLoss) {
    // fwd(commitment_loss) = mse(q,x) + BETA*mse(q,x)
    *outLoss = (float)(*acc * (1.0 + BETA_F) / (double)((size_t)NROWS * DIMS));
}

// ---------------------------------------------------------------------------
extern "C" void kernel_launch(void* const* d_in, const int* in_sizes, int n_in,
                              void* d_out, int out_size, void* d_ws, size_t ws_size,
                              hipStream_t stream) {
    const float* x  = (const float*)d_in[0];   // [N, D]
    const float* cb = (const float*)d_in[1];   // [K, D]

    float* out      = (float*)d_out;
    float* outQ     = out;                         // N*D
    float* outLoss  = out + (size_t)NROWS * DIMS;  // 1
    float* outIdxF  = outLoss + 1;                 // N

    char*   ws    = (char*)d_ws;
    int*    idx   = (int*)ws;                                  // N ints
    double* acc   = (double*)(ws + (size_t)NROWS * sizeof(int));
    float*  enorm = (float*)(ws + (size_t)NROWS * sizeof(int) + 16); // K floats

    hipMemsetAsync(acc, 0, sizeof(double), stream);

    vq_norms_kernel<<<KCODES, 128, 0, stream>>>(cb, enorm);

    const int smemBytes = (M_TILE * LSTRIDE + 2 * N_TILE * LSTRIDE) * (int)sizeof(float)
                        + M_TILE * N_TILE * (int)(sizeof(float) + sizeof(int));
    hipFuncSetAttribute((const void*)vq_argmin_kernel,
                        hipFuncAttributeMaxDynamicSharedMemorySize, smemBytes);
    vq_argmin_kernel<<<NROWS / M_TILE, 256, smemBytes, stream>>>(x, cb, enorm, idx);

    vq_gather_kernel<<<NROWS, 128, 0, stream>>>(x, cb, idx, outQ, outIdxF, acc);
    vq_finalize_kernel<<<1, 1, 0, stream>>>(acc, outLoss);
}